// ButterflyTransform_73615739453552
// MI455X (gfx1250) — compile-verified
//
#include <hip/hip_runtime.h>
#include <stdint.h>

// Fused 12-layer butterfly, one HBM pass:
//   - async global<->LDS DMA (ASYNCcnt) stages 16 rows x 4096 f32 per WG
//   - each group of 4 layers == 256 dense 16x16 composite matrices
//     (precomputed into d_ws, shared by all rows), applied with
//     V_WMMA_F32_16X16X4_F32 on 16-row x 16-element tiles.

#define NN      4096
#define HALFN   2048
#define RPB     16
#define TPB     512
// +2 floats pad per 16 (8B-aligned, intra-row conflict-free) and +4 row skew
// so that the 16 batch rows map to distinct banks (4*m mod 64).
#define PADF(i) ((i) + ((((i) >> 4)) << 1))
#define ROWF    (NN + ((NN >> 4) << 1) + 4)     // 4612 floats per padded row
#define LDS_BYTES (RPB * ROWF * 4)              // 295168 B (<= 320KB WGP LDS)

#define NBLK  256                                // 16-element blocks per group
#define NGRP  3
#define WS_FLOATS (NGRP * NBLK * 4 * 32 * 2)     // 196608 f32 = 768KB

typedef float v2f __attribute__((ext_vector_type(2)));
typedef float v8f __attribute__((ext_vector_type(8)));

// ---- group views: element index of slot k in block c ----
//   G=0: layers 0..3  (stride 2048..256):  idx = k*256 + c
//   G=1: layers 4..7  (stride  128..16):   idx = (c>>4)*256 + k*16 + (c&15)
//   G=2: layers 8..11 (stride    8..1):    idx = c*16 + k
__device__ __forceinline__ int idx_rt(int G, int c, int k) {
  if (G == 0) return k * 256 + c;
  if (G == 1) return (c >> 4) * 256 + k * 16 + (c & 15);
  return c * 16 + k;
}
template <int G>
__device__ __forceinline__ int idx_of(int c, int k) {
  if constexpr (G == 0) return k * 256 + c;
  else if constexpr (G == 1) return (c >> 4) * 256 + k * 16 + (c & 15);
  else return c * 16 + k;
}

__device__ __forceinline__ uint32_t lds_addr32(const void* p) {
  return (uint32_t)(uintptr_t)p;   // low 32 bits = LDS byte offset (aperture rule)
}
__device__ __forceinline__ void async_gl2lds_b64(uint32_t lds_byte, const void* gptr) {
  asm volatile("global_load_async_to_lds_b64 %0, %1, off"
               :: "v"(lds_byte), "v"((uint64_t)(uintptr_t)gptr) : "memory");
}
__device__ __forceinline__ void async_lds2gl_b64(void* gptr, uint32_t lds_byte) {
  asm volatile("global_store_async_from_lds_b64 %0, %1, off"
               :: "v"((uint64_t)(uintptr_t)gptr), "v"(lds_byte) : "memory");
}
__device__ __forceinline__ void wait_async_zero() {
  asm volatile("s_wait_asynccnt 0" ::: "memory");
}

// ============================================================================
// Kernel 1: build composite 16x16 matrices M (4 butterfly layers per group),
// stored directly in WMMA A-fragment lane order:
//   ws[((G*256+c)*4 + kk)*32 + lane] (as float2) = M[lane&15][kk*4+(lane>>4)*2 + {0,1}]
// One thread per (G, c, column k): propagate basis vector e_k through 4 layers.
// ============================================================================
__global__ void __launch_bounds__(256)
build_composites(const float* __restrict__ tw, float* __restrict__ ws)
{
  const int gid = blockIdx.x * 256 + threadIdx.x;
  if (gid >= NGRP * NBLK * 16) return;
  const int k = gid & 15;
  const int c = (gid >> 4) & 255;
  const int G = gid >> 12;

  const float4* tw4 = (const float4*)tw;
  float v[16];
#pragma unroll
  for (int m = 0; m < 16; ++m) v[m] = (m == k) ? 1.f : 0.f;

#pragma unroll
  for (int l = 0; l < 4; ++l) {
    const int L  = G * 4 + l;
    const int d  = 8 >> l;                 // pair distance in k
    const int sh = 12 - L;                 // log2(2*stride)
    const int sm = (NN >> (L + 1)) - 1;    // stride-1
#pragma unroll
    for (int kt = 0; kt < 16; ++kt) {
      if ((kt & d) == 0) {
        const int itop = idx_rt(G, c, kt);
        const int j = ((itop >> sh) << (sh - 1)) + (itop & sm);
        const float4 w = tw4[L * HALFN + j];
        const float a = v[kt], b = v[kt + d];
        v[kt]     = w.x * a + w.y * b;
        v[kt + d] = w.z * a + w.w * b;
      }
    }
  }
  // v[m] = M[m][k]; scatter column k into A-fragment order
  const int kk   = k >> 2;
  const int half = (k >> 1) & 1;
  const int comp = k & 1;
  float* base = ws + ((size_t)((G * NBLK + c) * 4 + kk) * 32) * 2;
#pragma unroll
  for (int m = 0; m < 16; ++m)
    base[(half * 16 + m) * 2 + comp] = v[m];
}

// ============================================================================
// WMMA group stage:  D(16 out-el x 16 rows) = M(16x16) * in^T(16 x 16 rows)
// A = M  (from ws, fragment-ordered, coalesced float2 per lane)
// B = in^T from LDS: lane L -> row (L&15), K = kk*4 + (L>>4)*2 + {0,1}
// D VGPR r -> out-el r + (L>>4)*8, row (L&15)  -> stored back to LDS in place.
// ============================================================================
template <int G>
__device__ __forceinline__ void wmma_group(float* lds, const v2f* __restrict__ wsv,
                                           int wv, int lane)
{
  const int m  = lane & 15;          // batch row
  const int hi = lane >> 4;          // lane half
  const int eh = hi * 2;             // K sub-offset within a 4-chunk
  const int rowbase = m * ROWF;

#pragma unroll 1
  for (int ib = 0; ib < 16; ++ib) {
    const int c = wv * 16 + ib;      // each of 16 waves owns 16 blocks
    const v2f* afr = wsv + (size_t)((G * NBLK + c) * 4) * 32 + lane;
    v8f d = {};
#pragma unroll
    for (int kk = 0; kk < 4; ++kk) {
      const v2f a = afr[kk * 32];
      v2f b;
      b.x = lds[rowbase + PADF(idx_of<G>(c, kk * 4 + eh))];
      b.y = lds[rowbase + PADF(idx_of<G>(c, kk * 4 + eh + 1))];
      d = __builtin_amdgcn_wmma_f32_16x16x4_f32(
              /*neg_a=*/false, a, /*neg_b=*/false, b,
              /*c_mod=*/(short)0, d, /*reuse_a=*/false, /*reuse_b=*/false);
    }
#pragma unroll
    for (int r = 0; r < 8; ++r) {
      const int el = r + hi * 8;
      lds[rowbase + PADF(idx_of<G>(c, el))] = d[r];
    }
  }
}

// ============================================================================
// Kernel 2: fused transform. Stage 16 rows in, 3 WMMA group stages, stage out.
// ============================================================================
__global__ void __launch_bounds__(TPB)
butterfly_fused(const float* __restrict__ x,
                const float* __restrict__ ws,
                float* __restrict__ out)
{
  extern __shared__ float lds[];
  const int t    = threadIdx.x;
  const int row0 = blockIdx.x * RPB;
  const uint32_t lbase = lds_addr32(lds);

  // ---- Stage in: async DMA 16 rows x 4096 f32 (B64/lane, coalesced) ----
  {
    const float* gbase = x + (size_t)row0 * NN;
#pragma unroll 4
    for (int it = 0; it < (RPB * (NN / 2)) / TPB; ++it) {   // 64 iters
      const int f2 = it * TPB + t;
      const int r  = f2 >> 11;                 // 2048 float2 per row
      const int i  = (f2 & 2047) * 2;          // even -> pad-safe, 8B aligned
      async_gl2lds_b64(lbase + (uint32_t)(r * ROWF + PADF(i)) * 4u,
                       gbase + (size_t)r * NN + i);
    }
    wait_async_zero();
  }
  __syncthreads();

  const int wv   = t >> 5;
  const int lane = t & 31;
  const v2f* wsv = (const v2f*)ws;

  wmma_group<0>(lds, wsv, wv, lane);
  __syncthreads();
  wmma_group<1>(lds, wsv, wv, lane);
  __syncthreads();
  wmma_group<2>(lds, wsv, wv, lane);
  __syncthreads();

  // ---- Stage out: async DMA LDS -> global ----
  {
    float* gbase = out + (size_t)row0 * NN;
#pragma unroll 4
    for (int it = 0; it < (RPB * (NN / 2)) / TPB; ++it) {
      const int f2 = it * TPB + t;
      const int r  = f2 >> 11;
      const int i  = (f2 & 2047) * 2;
      async_lds2gl_b64(gbase + (size_t)r * NN + i,
                       lbase + (uint32_t)(r * ROWF + PADF(i)) * 4u);
    }
    wait_async_zero();
  }
}

extern "C" void kernel_launch(void* const* d_in, const int* in_sizes, int n_in,
                              void* d_out, int out_size, void* d_ws, size_t ws_size,
                              hipStream_t stream)
{
  const float* x  = (const float*)d_in[0];
  const float* tw = (const float*)d_in[1];
  float* out      = (float*)d_out;
  float* ws       = (float*)d_ws;          // needs 768KB (WS_FLOATS * 4)

  const int batch = in_sizes[0] / NN;      // 8192
  const int grid  = batch / RPB;           // 512 workgroups

  // 1) composite 16x16 matrices per group/block (tiny, L2-resident result)
  build_composites<<<(NGRP * NBLK * 16 + 255) / 256, 256, 0, stream>>>(tw, ws);

  // 2) fused transform, 288KB dynamic LDS per workgroup
  (void)hipFuncSetAttribute((const void*)butterfly_fused,
                            hipFuncAttributeMaxDynamicSharedMemorySize,
                            LDS_BYTES);
  butterfly_fused<<<grid, TPB, LDS_BYTES, stream>>>(x, ws, out);
}